// FeatureShader_10033043603814
// MI455X (gfx1250) — compile-verified
//
#include <hip/hip_runtime.h>
#include <cstdint>

// ---- problem constants (from reference) ----
constexpr int KFR   = 8;              // fragments per pixel
constexpr int CH    = 16;             // feature channels
constexpr int NPIX  = 4 * 256 * 256;  // N*H*W
constexpr float ZFAR  = 100.0f;
constexpr float ZNEAR = 1.0f;
constexpr float INV_ZRANGE = 1.0f / (ZFAR - ZNEAR);
constexpr float INV_SIGMA  = 1.0f / 1e-4f;
constexpr float INV_GAMMA  = 1.0f / 1e-4f;
constexpr float EPSV       = 1e-10f;

constexpr int PPB     = 16;        // pixels per block
constexpr int THREADS = PPB * CH;  // 256 threads = 8 wave32

// LDS staging layout (bytes):
//   [0,512)     pix_to_face  : 128 ints    (16 px * 8 frag)
//   [512,1024)  dists        : 128 floats
//   [1024,1536) zbuf         : 128 floats
//   [1536,3072) bary         : 384 floats  (16 px * 8 frag * 3)
constexpr int LDS_BYTES = 3072;

__global__ __launch_bounds__(THREADS)
void feature_shader_kernel(const float* __restrict__ dists,
                           const float* __restrict__ zbuf,
                           const float* __restrict__ bary,
                           const float* __restrict__ verts,   // (V, C)
                           const float* __restrict__ bg,      // (C,)
                           const int*   __restrict__ p2f,
                           const int*   __restrict__ faces,   // (F, 3)
                           float*       __restrict__ out)     // (NPIX, C)
{
    __shared__ __align__(16) unsigned char smem[LDS_BYTES];

    const int t    = threadIdx.x;
    const int lane = t & 31;
    // Provably wave-uniform wave id -> scalar (s_cmp/s_cbranch) staging
    // branches with full EXEC on every async issue.
    const int wave = __builtin_amdgcn_readfirstlane(t >> 5);   // 0..7
    const int P0   = blockIdx.x * PPB;                         // first pixel

    // ---------------------------------------------------------------
    // Async global->LDS staging (ASYNCcnt path). One region per wave.
    // 6 waves x 32 lanes x 16B = 3072B.
    // ---------------------------------------------------------------
    if (wave < 6) {
        const void* gsrc;
        unsigned    off;
        if (wave == 0) {          // pix_to_face: 512B
            gsrc = (const void*)(p2f   + (P0 * KFR     + lane * 4));
            off  = 0u + (unsigned)lane * 16u;
        } else if (wave == 1) {   // dists: 512B
            gsrc = (const void*)(dists + (P0 * KFR     + lane * 4));
            off  = 512u + (unsigned)lane * 16u;
        } else if (wave == 2) {   // zbuf: 512B
            gsrc = (const void*)(zbuf  + (P0 * KFR     + lane * 4));
            off  = 1024u + (unsigned)lane * 16u;
        } else {                  // bary: 1536B across waves 3..5
            const int u = (wave - 3) * 32 + lane;
            gsrc = (const void*)(bary  + (P0 * KFR * 3 + u * 4));
            off  = 1536u + (unsigned)u * 16u;
        }
        // Generic pointer to a __shared__ object: low 32 bits == LDS byte addr.
        unsigned lds_addr = (unsigned)(uintptr_t)(void*)smem + off;
        unsigned long long gaddr = (unsigned long long)(uintptr_t)gsrc;
        asm volatile("global_load_async_to_lds_b128 %0, %1, off"
                     :: "v"(lds_addr), "v"(gaddr)
                     : "memory");
    }
    asm volatile("s_wait_asynccnt 0" ::: "memory");
    __syncthreads();

    // ---------------------------------------------------------------
    // Compute: 16 lanes per pixel, lane = channel. All per-pixel
    // fragment state pulled from LDS with 12 ds_load_b128 per thread.
    // ---------------------------------------------------------------
    const int p = t >> 4;      // local pixel 0..15
    const int c = t & 15;      // channel 0..15

    const int*   s_p2f  = (const int*)  (smem)          + p * KFR;
    const float* s_dist = (const float*)(smem + 512)    + p * KFR;
    const float* s_z    = (const float*)(smem + 1024)   + p * KFR;
    const float* s_bary = (const float*)(smem + 1536)   + p * KFR * 3;

    int   fs[KFR];
    float ds_[KFR];
    float zs[KFR];
    float bs[KFR * 3];
#pragma unroll
    for (int i = 0; i < 2; ++i) {
        ((int4*)fs)[i]    = ((const int4*)  s_p2f)[i];   // 32B, 16B-aligned
        ((float4*)ds_)[i] = ((const float4*)s_dist)[i];
        ((float4*)zs)[i]  = ((const float4*)s_z)[i];
    }
#pragma unroll
    for (int i = 0; i < 6; ++i) {
        ((float4*)bs)[i]  = ((const float4*)s_bary)[i];  // 96B, 16B-aligned
    }

    // Pass 1: masked z_inv per fragment + running max (clipped at EPS).
    float zinv[KFR];
    float zmax = EPSV;
#pragma unroll
    for (int k = 0; k < KFR; ++k) {
        const float zi = (fs[k] >= 0) ? (ZFAR - zs[k]) * INV_ZRANGE : 0.0f;
        zinv[k] = zi;
        zmax    = fmaxf(zmax, zi);
    }

    // Pass 2: weights + barycentric texel blend, accumulated per channel.
    float acc  = 0.0f;
    float wsum = 0.0f;
#pragma unroll
    for (int k = 0; k < KFR; ++k) {
        const int f = fs[k];
        if (f >= 0) {
            const float prob = 1.0f / (1.0f + __expf(ds_[k] * INV_SIGMA)); // sigmoid(-d/sigma)
            const float w    = prob * __expf((zinv[k] - zmax) * INV_GAMMA);
            wsum += w;

            const int v0 = faces[3 * f + 0];
            const int v1 = faces[3 * f + 1];
            const int v2 = faces[3 * f + 2];

            // 16 lanes of this pixel hit 16 consecutive dwords -> 64B line.
            float tex = bs[3 * k + 0] * verts[v0 * CH + c];
            tex = fmaf(bs[3 * k + 1], verts[v1 * CH + c], tex);
            tex = fmaf(bs[3 * k + 2], verts[v2 * CH + c], tex);

            acc = fmaf(w, tex, acc);
        }
    }

    const float delta = fmaxf(__expf((EPSV - zmax) * INV_GAMMA), EPSV);
    const float denom = wsum + delta;
    const float color = fmaf(delta, bg[c], acc) / denom;

    out[(P0 + p) * CH + c] = color;   // contiguous 1KB store per block
}

extern "C" void kernel_launch(void* const* d_in, const int* in_sizes, int n_in,
                              void* d_out, int out_size, void* d_ws, size_t ws_size,
                              hipStream_t stream) {
    const float* dists = (const float*)d_in[0];
    const float* zbuf  = (const float*)d_in[1];
    const float* bary  = (const float*)d_in[2];
    const float* verts = (const float*)d_in[3];
    const float* bg    = (const float*)d_in[4];
    const int*   p2f   = (const int*)  d_in[5];
    const int*   faces = (const int*)  d_in[6];
    float*       out   = (float*)d_out;

    dim3 grid(NPIX / PPB);
    dim3 block(THREADS);
    hipLaunchKernelGGL(feature_shader_kernel, grid, block, 0, stream,
                       dists, zbuf, bary, verts, bg, p2f, faces, out);
}